// QLSTM_65481071400435
// MI455X (gfx1250) — compile-verified
//
#include <hip/hip_runtime.h>
#include <math.h>

typedef float v2f __attribute__((ext_vector_type(2)));
typedef float v8f __attribute__((ext_vector_type(8)));
typedef float f4  __attribute__((ext_vector_type(4)));

#define NQ      8
#define NSTATES 256
#define T_STEPS 512
#define BATCH   128
#define HID     1024
#define HID4    (HID / 4)   // 256 float4 per H-row

struct GateParams {
    const float* head_w;  // (HID, NQ) row-major
    const float* head_b;  // (HID,)
    const float* rnd;     // (20,)
    const float* rx;      // scalar
    const float* ry;      // scalar
};
struct AllGates { GateParams g[4]; };

// ---- statevector gate helpers (state in LDS, one amplitude per thread) ----
// wire w lives at flat-index bit position (7 - w) for C-order flatten of (2,)*8.

__device__ __forceinline__ void apply_ry(float* sre, float* simg, int tid, float theta, int w) {
    int p = 7 - w, mask = 1 << p, bit = (tid >> p) & 1, partner = tid ^ mask;
    float c = cosf(0.5f * theta), s = sinf(0.5f * theta);
    float orr = sre[tid], oi = simg[tid], pr = sre[partner], pi = simg[partner];
    __syncthreads();
    float nr, ni;
    if (bit == 0) { nr = c * orr - s * pr; ni = c * oi - s * pi; }   // U[0,*] = [c, -s]
    else          { nr = c * orr + s * pr; ni = c * oi + s * pi; }   // U[1,*] = [s,  c]
    sre[tid] = nr; simg[tid] = ni;
    __syncthreads();
}

__device__ __forceinline__ void apply_rx(float* sre, float* simg, int tid, float theta, int w) {
    int p = 7 - w, mask = 1 << p, partner = tid ^ mask;
    float c = cosf(0.5f * theta), s = sinf(0.5f * theta);
    float orr = sre[tid], oi = simg[tid], pr = sre[partner], pi = simg[partner];
    __syncthreads();
    // new = c*own + (-i*s)*partner  (off-diagonal identical for both bit values)
    sre[tid] = c * orr + s * pi;
    simg[tid] = c * oi - s * pr;
    __syncthreads();
}

__device__ __forceinline__ void apply_cnot(float* sre, float* simg, int tid, int cw, int tw) {
    int pc = 7 - cw, pt = 7 - tw;
    int src = ((tid >> pc) & 1) ? (tid ^ (1 << pt)) : tid;
    float r = sre[src], i = simg[src];
    __syncthreads();
    sre[tid] = r; simg[tid] = i;
    __syncthreads();
}

// ---- kernel 1: quantum sim + WMMA head matvec + activation, one block per gate ----
__global__ void __launch_bounds__(256) qgate_kernel(AllGates P, float* ws_gates) {
    __shared__ float sre[NSTATES], simg[NSTATES], prob[NSTATES], zsh[NQ];
    const int tid  = threadIdx.x;
    const int gate = blockIdx.x;
    const GateParams gp = P.g[gate];

    sre[tid]  = (tid == 0) ? 1.0f : 0.0f;
    simg[tid] = 0.0f;
    __syncthreads();

    // fixed 20-op pseudo-random layer: even k -> RY(rand[k]) on wire k%8,
    //                                  odd  k -> CNOT(k%8, (k%8+1)%8)
    for (int k = 0; k < 20; ++k) {
        int w = k & 7;
        if ((k & 1) == 0) apply_ry(sre, simg, tid, gp.rnd[k], w);
        else              apply_cnot(sre, simg, tid, w, (w + 1) & 7);
    }
    float rxv = gp.rx[0], ryv = gp.ry[0];
    for (int w = 0; w < NQ; ++w) {
        apply_rx(sre, simg, tid, rxv, w);
        apply_ry(sre, simg, tid, ryv, w);
    }

    // <PauliZ> per wire
    float re = sre[tid], im = simg[tid];
    prob[tid] = re * re + im * im;
    __syncthreads();
    if (tid < NQ) {
        int p = 7 - tid;
        float z = 0.0f;
        for (int s = 0; s < NSTATES; ++s)
            z += ((s >> p) & 1) ? -prob[s] : prob[s];
        zsh[tid] = z;
    }
    __syncthreads();

    // head matvec: out[n] = sum_k z[k] * head_w[n][k] + head_b[n]
    // via V_WMMA_F32_16X16X4_F32 (exact fp32). z in row 0 of A; head_w^T as B.
    const int wave = tid >> 5, lane = tid & 31;
    const int half = lane >> 4;      // 0: lanes 0-15 (K base +0), 1: lanes 16-31 (K base +2)
    const int n16  = lane & 15;
    const float* hw = gp.head_w;

    for (int tile = wave * 8; tile < wave * 8 + 8; ++tile) {  // 64 tiles of 16 cols over 8 waves
        const int nb = tile * 16;
        v8f acc = {};
#pragma unroll
        for (int k0 = 0; k0 < 8; k0 += 4) {
            const int kb = k0 + half * 2;
            v2f a, b;
            // A 16x4 f32 layout: VGPR0 = A[M=lane&15][K = half?2:0], VGPR1 = K+1. Row 0 = z.
            a.x = (n16 == 0) ? zsh[kb]     : 0.0f;
            a.y = (n16 == 0) ? zsh[kb + 1] : 0.0f;
            // B 4x16 f32 layout (mirror): VGPR0 = B[K = half?2:0][N=n16], VGPR1 = K+1.
            b.x = hw[(nb + n16) * NQ + kb];
            b.y = hw[(nb + n16) * NQ + kb + 1];
            acc = __builtin_amdgcn_wmma_f32_16x16x4_f32(
                /*neg_a=*/false, a, /*neg_b=*/false, b,
                /*c_mod=*/(short)0, acc, /*reuse_a=*/false, /*reuse_b=*/false);
        }
        if (lane < 16) {  // C/D VGPR0, lanes 0-15 hold M=0 row
            float v = acc[0] + gp.head_b[nb + lane];
            v = (gate == 2) ? tanhf(v) : 1.0f / (1.0f + expf(-v));  // update->tanh, else sigmoid
            ws_gates[gate * HID + nb + lane] = v;
        }
    }
}

// ---- kernel 2: batch-independent recurrence, build hx table (T,H) + cx_final vector ----
__global__ void __launch_bounds__(256) recur_kernel(const float* __restrict__ ws_gates,
                                                    float* __restrict__ ws_hx,
                                                    float* __restrict__ ws_cx) {
    const int h = blockIdx.x * 256 + threadIdx.x;  // 0..1023
    const float f  = ws_gates[0 * HID + h];
    const float iv = ws_gates[1 * HID + h];
    const float g  = ws_gates[2 * HID + h];
    const float o  = ws_gates[3 * HID + h];
    const float ig = iv * g;
    float cx = 0.0f;
    for (int t = 0; t < T_STEPS; ++t) {
        cx = fmaf(f, cx, ig);
        ws_hx[t * HID + h] = o * tanhf(cx);
    }
    ws_cx[h] = cx;
}

// ---- kernel 3: HBM-bound broadcast of (T,H) table over B -> stacked (T,B,H) ----
__global__ void __launch_bounds__(256) bcast_kernel(const f4* __restrict__ ws_hx4,
                                                    float* __restrict__ out) {
    const int t   = blockIdx.x;          // 0..511
    const int bc  = blockIdx.y;          // 0..7 (chunks of 16 batch rows)
    const int tid = threadIdx.x;         // 0..255, one float4 each -> full H row
    const f4 v = ws_hx4[t * HID4 + tid];
    f4* out4 = (f4*)out;
    const size_t rowbase = (size_t)t * BATCH * HID4;
#pragma unroll 4
    for (int bb = 0; bb < 16; ++bb) {
        const size_t b = (size_t)bc * 16 + bb;
        __builtin_nontemporal_store(v, &out4[rowbase + b * HID4 + tid]);
    }
}

// ---- kernel 4: hx_final = stacked[T-1] broadcast, cx_final = cx vector broadcast ----
__global__ void __launch_bounds__(256) finals_kernel(const f4* __restrict__ ws_hx4,
                                                     const f4* __restrict__ ws_cx4,
                                                     float* __restrict__ out) {
    const int b = blockIdx.x;            // 0..127
    const int tid = threadIdx.x;         // 0..255
    f4* out4 = (f4*)out;
    const size_t base1 = (size_t)T_STEPS * BATCH * HID4;          // hx_final
    const size_t base2 = base1 + (size_t)BATCH * HID4;            // cx_final
    __builtin_nontemporal_store(ws_hx4[(T_STEPS - 1) * HID4 + tid],
                                &out4[base1 + (size_t)b * HID4 + tid]);
    __builtin_nontemporal_store(ws_cx4[tid],
                                &out4[base2 + (size_t)b * HID4 + tid]);
}

extern "C" void kernel_launch(void* const* d_in, const int* in_sizes, int n_in,
                              void* d_out, int out_size, void* d_ws, size_t ws_size,
                              hipStream_t stream) {
    // d_in layout (setup_inputs dict order):
    // [0] inputs (dead code in reference), then per gate {forget,input,update,output}:
    // lin_w, lin_b, head_w, head_b, rand, rx, ry  at 1 + 7*g + {0..6}
    AllGates P;
    for (int g = 0; g < 4; ++g) {
        const int base = 1 + g * 7;
        P.g[g].head_w = (const float*)d_in[base + 2];
        P.g[g].head_b = (const float*)d_in[base + 3];
        P.g[g].rnd    = (const float*)d_in[base + 4];
        P.g[g].rx     = (const float*)d_in[base + 5];
        P.g[g].ry     = (const float*)d_in[base + 6];
    }

    float* ws       = (float*)d_ws;
    float* ws_gates = ws;                          // 4*1024 floats
    float* ws_hx    = ws + 4 * HID;                // 512*1024 floats (2 MB, L2-resident)
    float* ws_cx    = ws_hx + T_STEPS * HID;       // 1024 floats

    qgate_kernel<<<4, 256, 0, stream>>>(P, ws_gates);
    recur_kernel<<<4, 256, 0, stream>>>(ws_gates, ws_hx, ws_cx);
    bcast_kernel<<<dim3(T_STEPS, 8), 256, 0, stream>>>((const f4*)ws_hx, (float*)d_out);
    finals_kernel<<<BATCH, 256, 0, stream>>>((const f4*)ws_hx, (const f4*)ws_cx, (float*)d_out);
}